// DiffEMA_1202590843032
// MI455X (gfx1250) — compile-verified
//
#include <hip/hip_runtime.h>

// ============================================================================
// Truncated EMA conv (K=700 taps) as an exact first-order affine scan.
//   y[t] = r*y[t-1] + a*x[t] - a*r^K * x[max(t-K,0)],  y[-1] = x0*(1 - r^K)
// 3-phase scan; phase 3 does intra-tile scans as 16x16 triangular matmuls on
// the CDNA5 WMMA unit (V_WMMA_F32_16X16X4_F32, exact fp32).
// Memory-bound: ~16MB read (+ L2-resident shifted re-read) + 16MB write.
// ============================================================================

typedef float v2f __attribute__((ext_vector_type(2)));
typedef float v8f __attribute__((ext_vector_type(8)));

#define T_LEN   4194304
#define KTAPS   700
#define W_ELEMS 1024                       // elements per wave
#define N_WAVES (T_LEN / W_ELEMS)          // 4096
#define WPB     8                          // waves per block (256 thr)
#define N_BLK   (N_WAVES / WPB)            // 512

__device__ __forceinline__ float fpow_u(float b, int e) {
    float acc = 1.f, p = b;
    while (e > 0) { if (e & 1) acc *= p; p *= p; e >>= 1; }
    return acc;
}

__device__ __forceinline__ void ema_coeffs(const float* wA, float& a, float& r, float& aK) {
    float wa = wA[0];
    a  = 1.f / (1.f + __expf(-wa));
    r  = 1.f - a;
    aK = a * fpow_u(r, KTAPS);
}

// Load 8 floats of the tail stream x[e0-K .. e0-K+7] with left-replication pad.
// 700 % 4 == 0 and e0 % 8 == 0  =>  e0-700 is 16B-aligned: b128 fast path.
__device__ __forceinline__ void load_tail8(const float* __restrict__ x, int e0,
                                           float xm[8]) {
    if (e0 >= KTAPS) {
        float4 m0 = *(const float4*)(x + e0 - KTAPS);
        float4 m1 = *(const float4*)(x + e0 - KTAPS + 4);
        xm[0]=m0.x; xm[1]=m0.y; xm[2]=m0.z; xm[3]=m0.w;
        xm[4]=m1.x; xm[5]=m1.y; xm[6]=m1.z; xm[7]=m1.w;
    } else {
        #pragma unroll
        for (int q = 0; q < 8; ++q) {
            int mi = e0 + q - KTAPS; if (mi < 0) mi = 0;   // == replicated pad
            xm[q] = x[mi];
        }
    }
}

// ---------------------------------------------------------------------------
// Phase 1: per-wave decayed partial P_w = sum_{j in chunk} r^(last-j) * z[j]
// ---------------------------------------------------------------------------
__global__ __launch_bounds__(256, 2)
void ema_partials_kernel(const float* __restrict__ x, const float* __restrict__ wA,
                         float* __restrict__ P) {
    const int lane = threadIdx.x & 31;
    const int wave = blockIdx.x * WPB + (threadIdx.x >> 5);
    float a, r, aK; ema_coeffs(wA, a, r, aK);

    const int base = wave * W_ELEMS + lane * 32;   // 32 serial elements / lane
    float st = 0.f;
    #pragma unroll
    for (int j = 0; j < 32; j += 8) {
        float4 a0 = *(const float4*)(x + base + j);
        float4 a1 = *(const float4*)(x + base + j + 4);
        float xs[8] = {a0.x,a0.y,a0.z,a0.w, a1.x,a1.y,a1.z,a1.w};
        float xm[8];
        load_tail8(x, base + j, xm);
        #pragma unroll
        for (int q = 0; q < 8; ++q)
            st = r * st + (a * xs[q] - aK * xm[q]);
    }
    // weight lane-partials so they sum to the wave partial (decay to chunk end)
    const float r32 = fpow_u(r, 32);
    float v = st * fpow_u(r32, 31 - lane);
    #pragma unroll
    for (int m = 16; m >= 1; m >>= 1) v += __shfl_xor(v, m, 32);
    if (lane == 0) P[wave] = v;
}

// ---------------------------------------------------------------------------
// Phase 2: affine-map scan over N_WAVES partials -> incoming state per wave.
//   S_0 = x0*(1-r^K);  S_{w+1} = r^1024 * S_w + P_w
// ---------------------------------------------------------------------------
__global__ __launch_bounds__(256, 1)
void ema_scan_kernel(const float* __restrict__ x, const float* __restrict__ wA,
                     const float* __restrict__ P, float* __restrict__ S) {
    __shared__ float sm[256], sp[256];
    const int t = threadIdx.x;
    float a, r, aK; ema_coeffs(wA, a, r, aK);
    const float rC = fpow_u(r, W_ELEMS);
    const float y0 = x[0] * (1.f - fpow_u(r, KTAPS));

    float pl[16];
    #pragma unroll
    for (int j = 0; j < 16; ++j) pl[j] = P[t * 16 + j];

    // local composition of 16 affine maps  y -> rC*y + P_j
    float m = 1.f, p = 0.f;
    #pragma unroll
    for (int j = 0; j < 16; ++j) { p = rC * p + pl[j]; m *= rC; }

    sm[t] = m; sp[t] = p; __syncthreads();
    float mAcc = m, pAcc = p;
    for (int off = 1; off < 256; off <<= 1) {          // Hillis-Steele on maps
        float qm = 1.f, qp = 0.f;
        if (t >= off) { qm = sm[t - off]; qp = sp[t - off]; }
        __syncthreads();
        if (t >= off) { pAcc = mAcc * qp + pAcc; mAcc = mAcc * qm;
                        sm[t] = mAcc; sp[t] = pAcc; }
        __syncthreads();
    }
    float Em = 1.f, Ep = 0.f;
    if (t > 0) { Em = sm[t - 1]; Ep = sp[t - 1]; }     // exclusive prefix

    float y = Em * y0 + Ep;                            // state before wave t*16
    #pragma unroll
    for (int j = 0; j < 16; ++j) { S[t * 16 + j] = y; y = rC * y + pl[j]; }
}

// ---------------------------------------------------------------------------
// Phase 3: per-wave scan of 1024 elements as 4 WMMA tiles of 256.
//   Tile as Z[16 pos][16 seg]; Y = L·Z (+ d ⊗ s carry)  via f32 WMMA.
// ---------------------------------------------------------------------------
__global__ __launch_bounds__(256, 2)
void ema_apply_kernel(const float* __restrict__ x, const float* __restrict__ wA,
                      const float* __restrict__ S, float* __restrict__ out) {
    __shared__ float zb[WPB][272];     // 16 segs * pitch 17 (bank-conflict free)
    __shared__ float y15b[WPB][16];    // row-15 of L·Z  (segment decayed sums)

    const int lane = threadIdx.x & 31;
    const int wv   = threadIdx.x >> 5;
    const int wave = blockIdx.x * WPB + wv;
    const int n    = lane & 15;        // matrix row (A) / column (B,D)
    const int half = lane >> 4;

    float a, r, aK; ema_coeffs(wA, a, r, aK);
    const float r16    = fpow_u(r, 16);
    const float rinv   = 1.f / r;
    const float rm     = fpow_u(r, n);                 // r^n
    const float rinv2h = half ? (rinv * rinv) : 1.f;   // rinv^(2*half)
    const float rinv4  = fpow_u(rinv, 4);

    float st = S[wave];                // y[wave*1024 - 1]
    const int base = wave * W_ELEMS;

    for (int tile = 0; tile < 4; ++tile) {
        __syncthreads();               // zb/y15b reuse fence
        const int tb = base + tile * 256;

        // ---- z for 8 elements/lane -> LDS, layout [seg][pos] pitch 17 ----
        {
            const int e0 = tb + lane * 8;
            float4 xv0 = *(const float4*)(x + e0);
            float4 xv1 = *(const float4*)(x + e0 + 4);
            float xs[8] = {xv0.x, xv0.y, xv0.z, xv0.w, xv1.x, xv1.y, xv1.z, xv1.w};
            float xm[8];
            load_tail8(x, e0, xm);
            #pragma unroll
            for (int jj = 0; jj < 8; ++jj) {
                float z = a * xs[jj] - aK * xm[jj];
                int ei = lane * 8 + jj;                           // 0..255
                zb[wv][(ei >> 4) * 17 + (ei & 15)] = z;
            }
        }
        __syncthreads();

        // ---- Y = L @ Z : four V_WMMA_F32_16X16X4_F32 over K ----
        v8f c = {};
        float rv = rinv2h;             // rinv^j0 running value
        #pragma unroll
        for (int kk = 0; kk < 4; ++kk) {
            const int j0 = 4 * kk + 2 * half;
            v2f av, bv;
            // A = L: A[m][j] = r^(m-j) for j<=m else 0  (m = n here)
            av.x = (n >= j0    ) ? rm * rv        : 0.f;
            av.y = (n >= j0 + 1) ? rm * rv * rinv : 0.f;
            // B = Z chunk rows j0, j0+1 at column n
            bv.x = zb[wv][n * 17 + j0];
            bv.y = zb[wv][n * 17 + j0 + 1];
            c = __builtin_amdgcn_wmma_f32_16x16x4_f32(
                    false, av, false, bv, (short)0, c, false, false);
            rv *= rinv4;
        }

        // ---- per-segment carries: s_c = r16*s_{c-1} + Y15[c-1], s_0 = st ----
        if (half) y15b[wv][n] = c[7];          // D row 15 lives in lanes 16-31, VGPR7
        __syncthreads();
        float s = st;
        #pragma unroll
        for (int j = 0; j < 15; ++j) {
            float yj = y15b[wv][j];            // uniform LDS broadcast
            if (j < n) s = r16 * s + yj;
        }

        // ---- carry as rank-1 WMMA:  Y += d ⊗ s,  d[m] = r^(m+1) ----
        {
            v2f av, bv;
            av.x = half ? 0.f : rm * r;        // A column 0 = d, cols 1..3 = 0
            av.y = 0.f;
            bv.x = half ? 0.f : s;             // B row 0 = s, rows 1..3 = 0
            bv.y = 0.f;
            c = __builtin_amdgcn_wmma_f32_16x16x4_f32(
                    false, av, false, bv, (short)0, c, false, false);
        }

        st = __shfl(c[7], 31, 32);             // D[15][15] = last y of tile

        // ---- store: lane (n,half) holds rows half*8..+7 of column n ----
        float* op = out + tb + n * 16 + half * 8;
        *(float4*)(op)     = make_float4(c[0], c[1], c[2], c[3]);
        *(float4*)(op + 4) = make_float4(c[4], c[5], c[6], c[7]);
    }
}

// ---------------------------------------------------------------------------
extern "C" void kernel_launch(void* const* d_in, const int* in_sizes, int n_in,
                              void* d_out, int out_size, void* d_ws, size_t ws_size,
                              hipStream_t stream) {
    const float* x  = (const float*)d_in[0];
    const float* wA = (const float*)d_in[1];
    float* out = (float*)d_out;
    float* P = (float*)d_ws;          // N_WAVES floats
    float* Sst = P + N_WAVES;         // N_WAVES floats

    ema_partials_kernel<<<N_BLK, 256, 0, stream>>>(x, wA, P);
    ema_scan_kernel    <<<1,     256, 0, stream>>>(x, wA, P, Sst);
    ema_apply_kernel   <<<N_BLK, 256, 0, stream>>>(x, wA, Sst, out);
}